// SpectralAxialBlock_44461501448572
// MI455X (gfx1250) — compile-verified
//
#include <hip/hip_runtime.h>
#include <math.h>

typedef float v2f __attribute__((ext_vector_type(2)));
typedef float v8f __attribute__((ext_vector_type(8)));

#define NB 16
#define NC 64
#define NT 128
#define ND 256
#define NH 8
#define DK 32
#define NTOK (NB * NC * NT)   // 131072

// V_WMMA_F32_16X16X4_F32 : full fp32 precision, D = A(16x4) * B(4x16) + C
static __device__ __forceinline__ v8f wmma4(v2f a, v2f b, v8f c) {
  return __builtin_amdgcn_wmma_f32_16x16x4_f32(false, a, false, b, (short)0, c,
                                               false, false);
}

static __device__ __forceinline__ float wsum(float v) {
#pragma unroll
  for (int m = 16; m; m >>= 1) v += __shfl_xor(v, m, 32);
  return v;
}
static __device__ __forceinline__ float wmax(float v) {
#pragma unroll
  for (int m = 16; m; m >>= 1) v = fmaxf(v, __shfl_xor(v, m, 32));
  return v;
}

// ---------------------------------------------------------------------------
// Fused LayerNorm + QKV GEMM.  Block = 64 tokens, 256 threads (8 waves).
// Writes q,k,v into (3, seq, head, pos, dk) scratch layout for the attention
// kernel.  MODE 0: time axis (seq = b*C+c, pos = t).  MODE 1: channel axis
// (seq = b*T+t, pos = c).
// ---------------------------------------------------------------------------
template <int MODE>
__global__ __launch_bounds__(256) void ln_qkv_kernel(
    const float* __restrict__ xsrc, const float* __restrict__ gam,
    const float* __restrict__ bet, const float* __restrict__ W,
    const float* __restrict__ bias, float* __restrict__ qkv) {
  __shared__ float alds[64 * 260];  // padded stride: 260 % 64 == 4
  const int tid = threadIdx.x, wave = tid >> 5, lane = tid & 31;
  const int half = lane >> 4, l = lane & 15;
  const int base = blockIdx.x * 64;

  // LayerNorm: 8 tokens per wave
  for (int i = 0; i < 8; ++i) {
    int tl = wave * 8 + i;
    const float* row = xsrc + (size_t)(base + tl) * ND;
    float xv[8], s = 0.f, ss = 0.f;
#pragma unroll
    for (int j = 0; j < 8; ++j) {
      xv[j] = row[lane + j * 32];
      s += xv[j];
      ss += xv[j] * xv[j];
    }
    s = wsum(s);
    ss = wsum(ss);
    float mu = s * (1.f / ND);
    float rstd = rsqrtf(ss * (1.f / ND) - mu * mu + 1e-5f);
#pragma unroll
    for (int j = 0; j < 8; ++j) {
      int c = lane + j * 32;
      alds[tl * 260 + c] = (xv[j] - mu) * rstd * gam[c] + bet[c];
    }
  }
  __syncthreads();

  // GEMM: (64 x 256) @ (256 x 768); each wave owns 6 N-tiles, 4 M-subtiles.
  for (int nt = wave; nt < 48; nt += 8) {
    int ncol = nt * 16 + l;
    v8f acc[4] = {};
    for (int kb = 0; kb < ND; kb += 4) {
      int kk = kb + 2 * half;
      v2f bf;
      bf.x = W[kk * 768 + ncol];
      bf.y = W[(kk + 1) * 768 + ncol];
#pragma unroll
      for (int ms = 0; ms < 4; ++ms) {
        int m = ms * 16 + l;
        v2f af;
        af.x = alds[m * 260 + kk];
        af.y = alds[m * 260 + kk + 1];
        acc[ms] = wmma4(af, bf, acc[ms]);
      }
    }
    int s3 = ncol >> 8, h = (ncol >> 5) & 7, dj = ncol & 31;
    float bv = bias[ncol];
#pragma unroll
    for (int ms = 0; ms < 4; ++ms) {
#pragma unroll
      for (int i = 0; i < 8; ++i) {
        int m = ms * 16 + i + 8 * half;
        int tok = base + m;
        int seq, pos, L;
        if (MODE == 0) {
          seq = tok >> 7;
          pos = tok & 127;
          L = NT;
        } else {
          int b = tok >> 13, c = (tok >> 7) & 63, t = tok & 127;
          seq = b * NT + t;
          pos = c;
          L = NC;
        }
        size_t dst = (size_t)s3 * ((size_t)NTOK * ND) +
                     ((size_t)(seq * NH + h) * L + pos) * DK + dj;
        qkv[dst] = acc[ms][i] + bv;
      }
    }
  }
}

// ---------------------------------------------------------------------------
// Attention core: one workgroup per (sequence, head).  Entire head (q,k,v and
// the LxL score matrix) lives in LDS.  scores -> +rel_bias -> softmax -> @v.
// Writes output token-major (seq*L + pos, D) with the head's dk-slice.
// ---------------------------------------------------------------------------
template <int L>
__global__ __launch_bounds__(256) void attn_kernel(
    const float* __restrict__ qkv, const float* __restrict__ btab,
    float* __restrict__ aout, int offset) {
  constexpr int SS = L + 4;  // (L+4) % 64 == 4 -> conflict-free column walks
  constexpr int KS = 36;
  __shared__ float qld[L * KS];
  __shared__ float kld[L * KS];
  __shared__ float vld[L * KS];
  __shared__ float sld[L * SS];
  const int tid = threadIdx.x, wave = tid >> 5, lane = tid & 31;
  const int half = lane >> 4, l = lane & 15;
  const int sh = blockIdx.x;  // seq*8 + h
  const int h = sh & 7, seq = sh >> 3;
  const size_t qbase = (size_t)sh * L * DK;
  const size_t plane = (size_t)NTOK * ND;

  for (int idx = tid; idx < L * DK; idx += 256) {
    int r = idx >> 5, c = idx & 31;
    qld[r * KS + c] = qkv[qbase + idx];
    kld[r * KS + c] = qkv[plane + qbase + idx];
    vld[r * KS + c] = qkv[2 * plane + qbase + idx];
  }
  __syncthreads();

  constexpr int MT = L / 16;
  const float scale = 0.17677669529663687f;  // 1/sqrt(dk)
  for (int t = wave; t < MT * MT; t += 8) {
    int mt = t / MT, nt = t % MT;
    v8f acc = {};
#pragma unroll
    for (int kb = 0; kb < DK; kb += 4) {
      int kk = kb + 2 * half;
      v2f af;
      int m = mt * 16 + l;
      af.x = qld[m * KS + kk];
      af.y = qld[m * KS + kk + 1];
      v2f bf;  // B[kk][n] = k[n][kk]  (q @ k^T)
      int n = nt * 16 + l;
      bf.x = kld[n * KS + kk];
      bf.y = kld[n * KS + kk + 1];
      acc = wmma4(af, bf, acc);
    }
    int n = nt * 16 + l;
#pragma unroll
    for (int i = 0; i < 8; ++i) {
      int m = mt * 16 + i + 8 * half;
      sld[m * SS + n] = acc[i] * scale + btab[(n - m + offset) * NH + h];
    }
  }
  __syncthreads();

  // row softmax
  for (int r = wave; r < L; r += 8) {
    float vv[L / 32];
    float mx = -3.4e38f;
#pragma unroll
    for (int j = 0; j < L / 32; ++j) {
      vv[j] = sld[r * SS + lane + j * 32];
      mx = fmaxf(mx, vv[j]);
    }
    mx = wmax(mx);
    float s = 0.f;
#pragma unroll
    for (int j = 0; j < L / 32; ++j) {
      vv[j] = expf(vv[j] - mx);
      s += vv[j];
    }
    s = wsum(s);
    float inv = 1.f / s;
#pragma unroll
    for (int j = 0; j < L / 32; ++j) sld[r * SS + lane + j * 32] = vv[j] * inv;
  }
  __syncthreads();

  // out = S @ V   (M=L, K=L, N=32)
  for (int t = wave; t < MT * 2; t += 8) {
    int mt = t >> 1, nt = t & 1;
    v8f acc = {};
    for (int kb = 0; kb < L; kb += 4) {
      int kk = kb + 2 * half;
      v2f af;
      int m = mt * 16 + l;
      af.x = sld[m * SS + kk];
      af.y = sld[m * SS + kk + 1];
      v2f bf;
      int n = nt * 16 + l;
      bf.x = vld[kk * KS + n];
      bf.y = vld[(kk + 1) * KS + n];
      acc = wmma4(af, bf, acc);
    }
    int n = nt * 16 + l;
#pragma unroll
    for (int i = 0; i < 8; ++i) {
      int m = mt * 16 + i + 8 * half;
      aout[((size_t)(seq * L + m)) * ND + h * DK + n] = acc[i];
    }
  }
}

// ---------------------------------------------------------------------------
// Output projection + residual.  Block = 64 seq-major tokens.
// MODE 0: identity token map.  MODE 1: (b,t,c) -> (b,c,t) scatter.
// ---------------------------------------------------------------------------
template <int MODE>
__global__ __launch_bounds__(256) void proj_res_kernel(
    const float* __restrict__ ain, const float* __restrict__ W,
    const float* __restrict__ bias, const float* __restrict__ xsrc,
    float* __restrict__ xdst) {
  __shared__ float alds[64 * 260];
  const int tid = threadIdx.x, wave = tid >> 5, lane = tid & 31;
  const int half = lane >> 4, l = lane & 15;
  const int base = blockIdx.x * 64;
  for (int idx = tid; idx < 64 * ND; idx += 256) {
    int r = idx >> 8, c = idx & 255;
    alds[r * 260 + c] = ain[(size_t)(base + r) * ND + c];
  }
  __syncthreads();
  for (int nt = wave; nt < 16; nt += 8) {
    int ncol = nt * 16 + l;
    v8f acc[4] = {};
    for (int kb = 0; kb < ND; kb += 4) {
      int kk = kb + 2 * half;
      v2f bf;
      bf.x = W[kk * ND + ncol];
      bf.y = W[(kk + 1) * ND + ncol];
#pragma unroll
      for (int ms = 0; ms < 4; ++ms) {
        int m = ms * 16 + l;
        v2f af;
        af.x = alds[m * 260 + kk];
        af.y = alds[m * 260 + kk + 1];
        acc[ms] = wmma4(af, bf, acc[ms]);
      }
    }
    float bv = bias[ncol];
#pragma unroll
    for (int ms = 0; ms < 4; ++ms)
#pragma unroll
      for (int i = 0; i < 8; ++i) {
        int m = ms * 16 + i + 8 * half;
        int tok = base + m;
        int xt;
        if (MODE == 0) {
          xt = tok;
        } else {
          int c = tok & 63, sq = tok >> 6, b = sq >> 7, t = sq & 127;
          xt = (b * NC + c) * NT + t;
        }
        size_t di = (size_t)xt * ND + ncol;
        xdst[di] = xsrc[di] + acc[ms][i] + bv;
      }
  }
}

// ---------------------------------------------------------------------------
// Fused LN + FFN (256 -> 1024 GELU -> 256) + residual.  Block = 16 tokens;
// the 16x1024 hidden tile stays in LDS (saves a 537 MB HBM round trip).
// ---------------------------------------------------------------------------
__global__ __launch_bounds__(256) void ffn_kernel(
    const float* __restrict__ x, const float* __restrict__ gam,
    const float* __restrict__ bet, const float* __restrict__ W1,
    const float* __restrict__ b1, const float* __restrict__ W2,
    const float* __restrict__ b2, float* __restrict__ out) {
  __shared__ float xln[16 * 260];
  __shared__ float hld[16 * 1028];  // 1028 % 64 == 4
  const int tid = threadIdx.x, wave = tid >> 5, lane = tid & 31;
  const int half = lane >> 4, l = lane & 15;
  const int base = blockIdx.x * 16;

  for (int i = 0; i < 2; ++i) {
    int tl = wave * 2 + i;
    const float* row = x + (size_t)(base + tl) * ND;
    float xv[8], s = 0.f, ss = 0.f;
#pragma unroll
    for (int j = 0; j < 8; ++j) {
      xv[j] = row[lane + j * 32];
      s += xv[j];
      ss += xv[j] * xv[j];
    }
    s = wsum(s);
    ss = wsum(ss);
    float mu = s * (1.f / ND);
    float rstd = rsqrtf(ss * (1.f / ND) - mu * mu + 1e-5f);
#pragma unroll
    for (int j = 0; j < 8; ++j) {
      int c = lane + j * 32;
      xln[tl * 260 + c] = (xv[j] - mu) * rstd * gam[c] + bet[c];
    }
  }
  __syncthreads();

  // h = gelu(xln @ W1 + b1)
  for (int nt = wave; nt < 64; nt += 8) {
    int ncol = nt * 16 + l;
    v8f acc = {};
    for (int kb = 0; kb < ND; kb += 4) {
      int kk = kb + 2 * half;
      v2f bf;
      bf.x = W1[kk * 1024 + ncol];
      bf.y = W1[(kk + 1) * 1024 + ncol];
      v2f af;
      af.x = xln[l * 260 + kk];
      af.y = xln[l * 260 + kk + 1];
      acc = wmma4(af, bf, acc);
    }
    float bv = b1[ncol];
#pragma unroll
    for (int i = 0; i < 8; ++i) {
      int m = i + 8 * half;
      float v = acc[i] + bv;
      hld[m * 1028 + ncol] = 0.5f * v * (1.f + erff(v * 0.7071067811865475f));
    }
  }
  __syncthreads();

  // out = x + h @ W2 + b2
  for (int nt = wave; nt < 16; nt += 8) {
    int ncol = nt * 16 + l;
    v8f acc = {};
    for (int kb = 0; kb < 1024; kb += 4) {
      int kk = kb + 2 * half;
      v2f bf;
      bf.x = W2[kk * ND + ncol];
      bf.y = W2[(kk + 1) * ND + ncol];
      v2f af;
      af.x = hld[l * 1028 + kk];
      af.y = hld[l * 1028 + kk + 1];
      acc = wmma4(af, bf, acc);
    }
    float bv = b2[ncol];
#pragma unroll
    for (int i = 0; i < 8; ++i) {
      int m = i + 8 * half;
      size_t di = (size_t)(base + m) * ND + ncol;
      out[di] = x[di] + acc[i] + bv;
    }
  }
}

// ---------------------------------------------------------------------------
extern "C" void kernel_launch(void* const* d_in, const int* in_sizes, int n_in,
                              void* d_out, int out_size, void* d_ws,
                              size_t ws_size, hipStream_t stream) {
  (void)in_sizes;
  (void)n_in;
  (void)out_size;
  (void)ws_size;
  const float* x       = (const float*)d_in[0];
  const float* ln_t_g  = (const float*)d_in[1];
  const float* ln_t_b  = (const float*)d_in[2];
  const float* t_qkv_w = (const float*)d_in[3];
  const float* t_qkv_b = (const float*)d_in[4];
  const float* t_pw    = (const float*)d_in[5];
  const float* t_pb    = (const float*)d_in[6];
  const float* t_bias  = (const float*)d_in[7];
  const float* ln_c_g  = (const float*)d_in[8];
  const float* ln_c_b  = (const float*)d_in[9];
  const float* c_qkv_w = (const float*)d_in[10];
  const float* c_qkv_b = (const float*)d_in[11];
  const float* c_pw    = (const float*)d_in[12];
  const float* c_pb    = (const float*)d_in[13];
  const float* c_bias  = (const float*)d_in[14];
  const float* ln_f_g  = (const float*)d_in[15];
  const float* ln_f_b  = (const float*)d_in[16];
  const float* w1      = (const float*)d_in[17];
  const float* fb1     = (const float*)d_in[18];
  const float* w2      = (const float*)d_in[19];
  const float* fb2     = (const float*)d_in[20];
  float* outp = (float*)d_out;

  float* ws_qkv = (float*)d_ws;                     // 3 * NTOK * ND f32
  float* ws_attn = ws_qkv + (size_t)3 * NTOK * ND;  // NTOK * ND f32
  float* ws_x = ws_attn + (size_t)NTOK * ND;        // NTOK * ND f32

  // ---- time attention ----
  ln_qkv_kernel<0><<<NTOK / 64, 256, 0, stream>>>(x, ln_t_g, ln_t_b, t_qkv_w,
                                                  t_qkv_b, ws_qkv);
  attn_kernel<128><<<(NB * NC) * NH, 256, 0, stream>>>(ws_qkv, t_bias, ws_attn,
                                                       127);
  proj_res_kernel<0><<<NTOK / 64, 256, 0, stream>>>(ws_attn, t_pw, t_pb, x,
                                                    ws_x);
  // ---- channel attention ----
  ln_qkv_kernel<1><<<NTOK / 64, 256, 0, stream>>>(ws_x, ln_c_g, ln_c_b,
                                                  c_qkv_w, c_qkv_b, ws_qkv);
  attn_kernel<64><<<(NB * NT) * NH, 256, 0, stream>>>(ws_qkv, c_bias, ws_attn,
                                                      63);
  proj_res_kernel<1><<<NTOK / 64, 256, 0, stream>>>(ws_attn, c_pw, c_pb, ws_x,
                                                    ws_x);
  // ---- FFN ----
  ffn_kernel<<<NTOK / 16, 256, 0, stream>>>(ws_x, ln_f_g, ln_f_b, w1, fb1, w2,
                                            fb2, outp);
}